// MotionEvaluator_1717986918756
// MI455X (gfx1250) — compile-verified
//
#include <hip/hip_runtime.h>
#include <stdint.h>

#define NJ        22
#define BLOCK     64
#define NELEM     (32 * 4096)
#define NBLOCKS   (NELEM / BLOCK)        // 2048
#define GT_F      132                    // floats per element in gt_rot (22*6)
#define PRED_F    60                     // floats per element in pred_pos (20*3)
#define RED_THREADS 256

// Kinematic tree (compile-time constant -> fully unrolled FK)
__device__ constexpr int PAR[NJ] =
    {-1, 0, 0, 0, 1, 2, 3, 4, 5, 6, 7, 8, 9, 9, 9, 12, 13, 14, 16, 17, 18, 19};

// ---- CDNA5 async global->LDS copy (ASYNCcnt path) -------------------------
typedef __attribute__((ext_vector_type(4))) int v4i;
typedef __attribute__((address_space(1))) v4i glob_v4i;
typedef __attribute__((address_space(3))) v4i lds_v4i;

__device__ __forceinline__ void async_copy_b128(const float* gsrc, float* ldst) {
#if defined(__AMDGCN__) && __has_builtin(__builtin_amdgcn_global_load_async_to_lds_b128)
    __builtin_amdgcn_global_load_async_to_lds_b128(
        (glob_v4i*)(gsrc),
        (lds_v4i*)(ldst),
        /*imm offset*/ 0, /*cpol*/ 0);
#else
    unsigned lds_addr =
        (unsigned)(uintptr_t)(__attribute__((address_space(3))) char*)(ldst);
    unsigned long long gaddr = (unsigned long long)(uintptr_t)gsrc;
    asm volatile("global_load_async_to_lds_b128 %0, %1, off"
                 :: "v"(lds_addr), "v"(gaddr)
                 : "memory");
#endif
}

__device__ __forceinline__ void wait_async0() {
#if defined(__AMDGCN__) && __has_builtin(__builtin_amdgcn_s_wait_asynccnt)
    __builtin_amdgcn_s_wait_asynccnt(0);
#else
    asm volatile("s_wait_asynccnt 0" ::: "memory");
#endif
}

// ---- main kernel: FK + per-element MPJPE partial sums ----------------------
__launch_bounds__(BLOCK)
__global__ void fk_mpjpe_kernel(const float* __restrict__ pred,   // [B,S,20,3]
                                const float* __restrict__ gt,     // [B,S,22,6]
                                const float* __restrict__ offs,   // [22,3]
                                float* __restrict__ partials)     // [NBLOCKS]
{
    __shared__ __align__(16) float s_gt[BLOCK * GT_F];      // 33792 B
    __shared__ __align__(16) float s_pred[BLOCK * PRED_F];  // 15360 B
    __shared__ float s_red[BLOCK];

    const int tid = threadIdx.x;
    const long long base_elem = (long long)blockIdx.x * BLOCK;
    const float* gbase = gt   + base_elem * GT_F;
    const float* pbase = pred + base_elem * PRED_F;

    // Stage 64 elements of gt_rot (528 B each) and pred_pos (240 B each) into
    // LDS with coalesced async b128: lanes within an iteration are contiguous,
    // so every issued instruction moves a dense 512 B/wave stream from HBM.
#pragma unroll
    for (int k = 0; k < GT_F / 4; ++k) {          // 33 chunks per thread
        const int c = tid + k * BLOCK;
        async_copy_b128(gbase + 4 * c, &s_gt[4 * c]);
    }
#pragma unroll
    for (int k = 0; k < PRED_F / 4; ++k) {        // 15 chunks per thread
        const int c = tid + k * BLOCK;
        async_copy_b128(pbase + 4 * c, &s_pred[4 * c]);
    }
    wait_async0();
    __syncthreads();

    const float* m  = &s_gt[tid * GT_F];       // this element's 22x6 rotations
    const float* pp = &s_pred[tid * PRED_F];   // this element's 20x3 predictions

    float R[NJ][9];   // global rotations (SROA'd to registers; short liveness)
    float P[NJ][3];   // global positions
    R[0][0] = 1.f; R[0][1] = 0.f; R[0][2] = 0.f;
    R[0][3] = 0.f; R[0][4] = 1.f; R[0][5] = 0.f;
    R[0][6] = 0.f; R[0][7] = 0.f; R[0][8] = 1.f;
    P[0][0] = 0.f; P[0][1] = 0.f; P[0][2] = 0.f;

    float sum = 0.f;

#pragma unroll
    for (int i = 1; i < NJ; ++i) {
        const int p = PAR[i];

        // g_pos[i] = g_pos[p] + g_rot[p] * offsets[i]   (uniform scalar loads)
        const float ox = offs[i * 3 + 0];
        const float oy = offs[i * 3 + 1];
        const float oz = offs[i * 3 + 2];
        P[i][0] = P[p][0] + R[p][0] * ox + R[p][1] * oy + R[p][2] * oz;
        P[i][1] = P[p][1] + R[p][3] * ox + R[p][4] * oy + R[p][5] * oz;
        P[i][2] = P[p][2] + R[p][6] * ox + R[p][7] * oy + R[p][8] * oz;

        // local rotation from 6D rep (identity for joint 1)
        float L[9];
        if (i == 1) {
            L[0] = 1.f; L[1] = 0.f; L[2] = 0.f;
            L[3] = 0.f; L[4] = 1.f; L[5] = 0.f;
            L[6] = 0.f; L[7] = 0.f; L[8] = 1.f;
        } else {
            const float a1x = m[i * 6 + 0], a1y = m[i * 6 + 1], a1z = m[i * 6 + 2];
            const float a2x = m[i * 6 + 3], a2y = m[i * 6 + 4], a2z = m[i * 6 + 5];
            const float n1 = fmaxf(sqrtf(a1x * a1x + a1y * a1y + a1z * a1z), 1e-12f);
            const float i1 = 1.0f / n1;
            const float b1x = a1x * i1, b1y = a1y * i1, b1z = a1z * i1;
            const float d  = b1x * a2x + b1y * a2y + b1z * a2z;
            float b2x = a2x - d * b1x, b2y = a2y - d * b1y, b2z = a2z - d * b1z;
            const float n2 = fmaxf(sqrtf(b2x * b2x + b2y * b2y + b2z * b2z), 1e-12f);
            const float i2 = 1.0f / n2;
            b2x *= i2; b2y *= i2; b2z *= i2;
            const float b3x = b1y * b2z - b1z * b2y;
            const float b3y = b1z * b2x - b1x * b2z;
            const float b3z = b1x * b2y - b1y * b2x;
            // columns are b1,b2,b3 (row-major storage)
            L[0] = b1x; L[1] = b2x; L[2] = b3x;
            L[3] = b1y; L[4] = b2y; L[5] = b3y;
            L[6] = b1z; L[7] = b2z; L[8] = b3z;
        }

        // g_rot[i] = g_rot[p] * L
#pragma unroll
        for (int r = 0; r < 3; ++r)
#pragma unroll
            for (int c = 0; c < 3; ++c)
                R[i][r * 3 + c] = R[p][r * 3 + 0] * L[0 + c]
                                + R[p][r * 3 + 1] * L[3 + c]
                                + R[p][r * 3 + 2] * L[6 + c];

        // distance vs predicted position (joints 2..21 -> pred index i-2)
        if (i >= 2) {
            const float dx = pp[(i - 2) * 3 + 0] - P[i][0];
            const float dy = pp[(i - 2) * 3 + 1] - P[i][1];
            const float dz = pp[(i - 2) * 3 + 2] - P[i][2];
            sum += sqrtf(dx * dx + dy * dy + dz * dz);
        }
    }

    // deterministic in-block reduction
    s_red[tid] = sum;
    __syncthreads();
#pragma unroll
    for (int w = BLOCK / 2; w > 0; w >>= 1) {
        if (tid < w) s_red[tid] += s_red[tid + w];
        __syncthreads();
    }
    if (tid == 0) partials[blockIdx.x] = s_red[0];
}

// ---- deterministic final reduction ----------------------------------------
__launch_bounds__(RED_THREADS)
__global__ void final_reduce_kernel(const float* __restrict__ partials,
                                    int n, float* __restrict__ out, float scale)
{
    __shared__ float s[RED_THREADS];
    float acc = 0.f;
    for (int i = threadIdx.x; i < n; i += RED_THREADS) acc += partials[i];
    s[threadIdx.x] = acc;
    __syncthreads();
#pragma unroll
    for (int w = RED_THREADS / 2; w > 0; w >>= 1) {
        if (threadIdx.x < w) s[threadIdx.x] += s[threadIdx.x + w];
        __syncthreads();
    }
    if (threadIdx.x == 0) out[0] = s[0] * scale;
}

extern "C" void kernel_launch(void* const* d_in, const int* in_sizes, int n_in,
                              void* d_out, int out_size, void* d_ws, size_t ws_size,
                              hipStream_t stream) {
    const float* pred = (const float*)d_in[0];   // [32,4096,20,3] f32
    const float* gt   = (const float*)d_in[1];   // [32,4096,22,6] f32
    const float* offs = (const float*)d_in[2];   // [22,3] f32
    float* out      = (float*)d_out;
    float* partials = (float*)d_ws;              // NBLOCKS floats (8 KB)

    fk_mpjpe_kernel<<<NBLOCKS, BLOCK, 0, stream>>>(pred, gt, offs, partials);

    const float scale = 1000.0f / ((float)NELEM * 20.0f);
    final_reduce_kernel<<<1, RED_THREADS, 0, stream>>>(partials, NBLOCKS, out, scale);
}